// SSIMLoss_16415365005830
// MI455X (gfx1250) — compile-verified
//
#include <hip/hip_runtime.h>

// SSIM loss on gfx1250 via V_WMMA_F32_16X16X4_F32 band-matrix convolution.
// pred/target: (16,3,512,512) f32. Output: one f32 scalar = 1 - mean(ssim_map).

typedef float v2f __attribute__((ext_vector_type(2)));
typedef float v8f __attribute__((ext_vector_type(8)));

#define IMG          512
#define PLANE_ELEMS  (IMG * IMG)
#define TILES_X      32
#define TILES_Y      32
#define N_PLANES     48                      // 16 batch * 3 channels
#define N_TILES      (TILES_X * TILES_Y * N_PLANES)   // 49152
#define WAVES_PB     4
#define N_BLOCKS     (N_TILES / WAVES_PB)    // 12288
#define SSIM_C1      1.0e-4f
#define SSIM_C2      9.0e-4f

// Normalized 11-tap Gaussian (sigma=1.5), padded with zeros so that a clamped
// index lookup implements the band function g[d] for d in (-inf, +inf).
__device__ __constant__ float G_TAB[16] = {
    0.00102838f, 0.00759876f, 0.03600077f, 0.10936070f, 0.21300530f,
    0.26601172f, 0.21300530f, 0.10936070f, 0.03600077f, 0.00759876f,
    0.00102838f, 0.0f, 0.0f, 0.0f, 0.0f, 0.0f};

__device__ __forceinline__ float band(int d) {
  unsigned u = (unsigned)d;            // negative -> huge -> clamps to 15 -> 0
  u = (u > 15u) ? 15u : u;             // v_min_u32
  return G_TAB[u];                     // cached global load (rodata)
}

__device__ __forceinline__ int iclamp(int v, int lo, int hi) {
  return v < lo ? lo : (v > hi ? hi : v);   // v_med3_i32
}

__device__ __forceinline__ v8f wmma_f32(v2f a, v2f b, v8f c) {
  // D = A(16x4 f32) * B(4x16 f32) + C(16x16 f32)
  return __builtin_amdgcn_wmma_f32_16x16x4_f32(
      /*neg_a=*/false, a, /*neg_b=*/false, b,
      /*c_mod=*/(short)0, c, /*reuse_a=*/false, /*reuse_b=*/false);
}

__global__ __launch_bounds__(WAVES_PB * 32)
void ssim_stage1(const float* __restrict__ pred,
                 const float* __restrict__ targ,
                 float* __restrict__ partial) {
  // Per-wave transposed h-buffer: [quantity][out_col n][local row 0..31]
  __shared__ float ht[WAVES_PB][5][16][32];
  __shared__ float wsum[WAVES_PB];

  const int tid  = threadIdx.x;
  const int lane = tid & 31;
  const int w    = tid >> 5;

  const int tile  = blockIdx.x * WAVES_PB + w;
  const int tx    = tile & (TILES_X - 1);
  const int ty    = (tile >> 5) & (TILES_Y - 1);
  const int plane = tile >> 10;

  // Wave-uniform 64-bit bases (SGPRs); per-lane addressing stays 32-bit so
  // loads lower to GVS mode: global_load_b32 v, v_off, s[base].
  const float* __restrict__ P = pred + (size_t)plane * PLANE_ELEMS;
  const float* __restrict__ T = targ + (size_t)plane * PLANE_ELEMS;

  const int n  = lane & 15;   // M (A rows) == N (B cols) == lane%16 in f32 WMMA layout
  const int hi = lane >> 4;   // K half-select
  const int x0 = tx * 16;
  const int y0 = ty * 16;

  // Band weights from the rodata table: B[k][n] = g[k-n]; by layout symmetry
  // the same registers serve as the vertical-pass A matrix.
  v2f bh[7];
#pragma unroll
  for (int kc = 0; kc < 7; ++kc) {
    const int k0 = kc * 4 + hi * 2;
    v2f b;
    b.x = band(k0 - n);
    b.y = band(k0 + 1 - n);
    bh[kc] = b;
  }

  // Hoisted per-row state for the two 16-row M blocks (rows y0-5 .. y0+26).
  int rowoff[2];
  bool rok[2];
#pragma unroll
  for (int mb = 0; mb < 2; ++mb) {
    const int row = y0 - 5 + mb * 16 + n;
    rok[mb] = (unsigned)row < (unsigned)IMG;
    rowoff[mb] = iclamp(row, 0, IMG - 1) << 9;   // 32-bit element offset
  }

  // ---- Horizontal pass ----
  v8f D[5][2] = {};
#pragma unroll
  for (int kc = 0; kc < 7; ++kc) {
    // Column clamps/masks shared across both M blocks and both images.
    const int c0 = x0 - 5 + kc * 4 + hi * 2;
    const int c1 = c0 + 1;
    const bool m0 = (unsigned)c0 < (unsigned)IMG;
    const bool m1 = (unsigned)c1 < (unsigned)IMG;
    const int cc0 = iclamp(c0, 0, IMG - 1);
    const int cc1 = iclamp(c1, 0, IMG - 1);
#pragma unroll
    for (int mb = 0; mb < 2; ++mb) {
      const int off0 = rowoff[mb] + cc0;   // 32-bit VGPR offsets, uniform SGPR base
      const int off1 = rowoff[mb] + cc1;
      const float p0 = P[off0];
      const float p1 = P[off1];
      const float t0 = T[off0];
      const float t1 = T[off1];
      const bool k0ok = rok[mb] && m0;
      const bool k1ok = rok[mb] && m1;
      v2f ap, at;
      ap.x = k0ok ? p0 : 0.0f;
      ap.y = k1ok ? p1 : 0.0f;
      at.x = k0ok ? t0 : 0.0f;
      at.y = k1ok ? t1 : 0.0f;
      const v2f app = ap * ap;
      const v2f att = at * at;
      const v2f apt = ap * at;
      D[0][mb] = wmma_f32(ap,  bh[kc], D[0][mb]);   // mu1 pre-h
      D[1][mb] = wmma_f32(at,  bh[kc], D[1][mb]);   // mu2 pre-h
      D[2][mb] = wmma_f32(app, bh[kc], D[2][mb]);   // E[p^2] pre-h
      D[3][mb] = wmma_f32(att, bh[kc], D[3][mb]);   // E[t^2] pre-h
      D[4][mb] = wmma_f32(apt, bh[kc], D[4][mb]);   // E[pt]  pre-h
    }
  }

  // Store h transposed: ht[w][q][n][local_row]; D vgpr r -> m = r + 8*hi.
#pragma unroll
  for (int q = 0; q < 5; ++q) {
#pragma unroll
    for (int mb = 0; mb < 2; ++mb) {
      float* colp = &ht[w][q][n][mb * 16 + hi * 8];
#pragma unroll
      for (int r = 0; r < 8; r += 2) {
        v2f v;
        v.x = D[q][mb][r];
        v.y = D[q][mb][r + 1];
        *(v2f*)(colp + r) = v;          // ds_store_b64, even-aligned
      }
    }
  }
  __syncthreads();

  // ---- Vertical pass: out[m][n] = sum_k g[k-m] * h[k][n] ----
  v8f E[5] = {};
#pragma unroll
  for (int kc = 0; kc < 7; ++kc) {
    const int k0 = kc * 4 + hi * 2;
#pragma unroll
    for (int q = 0; q < 5; ++q) {
      const v2f bv = *(const v2f*)(&ht[w][q][n][k0]);   // ds_load_b64
      E[q] = wmma_f32(bh[kc], bv, E[q]);
    }
  }

  // ---- SSIM map + reduction (8 output pixels per lane) ----
  float acc = 0.0f;
#pragma unroll
  for (int r = 0; r < 8; ++r) {
    const float mu1   = E[0][r];
    const float mu2   = E[1][r];
    const float mu1s  = mu1 * mu1;
    const float mu2s  = mu2 * mu2;
    const float mu12  = mu1 * mu2;
    const float s1    = E[2][r] - mu1s;
    const float s2    = E[3][r] - mu2s;
    const float s12   = E[4][r] - mu12;
    const float num   = (2.0f * mu12 + SSIM_C1) * (2.0f * s12 + SSIM_C2);
    const float den   = (mu1s + mu2s + SSIM_C1) * (s1 + s2 + SSIM_C2);
    acc += num / den;
  }
#pragma unroll
  for (int off = 16; off > 0; off >>= 1)
    acc += __shfl_xor(acc, off, 32);
  if (lane == 0) wsum[w] = acc;
  __syncthreads();
  if (tid == 0)
    partial[blockIdx.x] = (wsum[0] + wsum[1]) + (wsum[2] + wsum[3]);
}

__global__ __launch_bounds__(256)
void ssim_stage2(const float* __restrict__ partial, float* __restrict__ out) {
  __shared__ double sm[256];
  double a = 0.0;
  for (int i = threadIdx.x; i < N_BLOCKS; i += 256)
    a += (double)partial[i];
  sm[threadIdx.x] = a;
  __syncthreads();
  for (int o = 128; o > 0; o >>= 1) {
    if ((int)threadIdx.x < o) sm[threadIdx.x] += sm[threadIdx.x + o];
    __syncthreads();
  }
  if (threadIdx.x == 0) {
    const double mean = sm[0] / (double)(16.0 * 3.0 * 512.0 * 512.0);
    out[0] = (float)(1.0 - mean);   // LOSS_WEIGHT == 1
  }
}

extern "C" void kernel_launch(void* const* d_in, const int* in_sizes, int n_in,
                              void* d_out, int out_size, void* d_ws, size_t ws_size,
                              hipStream_t stream) {
  (void)in_sizes; (void)n_in; (void)out_size; (void)ws_size;
  const float* pred = (const float*)d_in[0];
  const float* targ = (const float*)d_in[1];
  float* partial    = (float*)d_ws;     // N_BLOCKS floats; fully overwritten each call
  float* out        = (float*)d_out;

  ssim_stage1<<<N_BLOCKS, WAVES_PB * 32, 0, stream>>>(pred, targ, partial);
  ssim_stage2<<<1, 256, 0, stream>>>(partial, out);
}